// CoulombLayer_68728066671213
// MI455X (gfx1250) — compile-verified
//
#include <hip/hip_runtime.h>

// ---------------------------------------------------------------------------
// CoulombLayer for MI455X (gfx1250, wave32).
// Memory-bound scatter workload: 192 MB streamed edge data @ 23.3 TB/s ~ 8 us
// floor. Edge streams are staged into LDS with CDNA5 async-to-LDS DMA loads
// (GLOBAL_LOAD_ASYNC_TO_LDS_B128, ASYNCcnt, double-buffered per wave), which
// frees the vector-load path for the random qi_c gathers. Scatter uses
// fire-and-forget global_atomic_add_f32 (relaxed, agent scope).
// ---------------------------------------------------------------------------

typedef float v4f __attribute__((ext_vector_type(4)));
typedef int   v4i __attribute__((ext_vector_type(4)));

#define CUTOFF_F 10.0f

// chi(r) = f(2r)*phi + (1-f(2r))/r,  phi = 1/sqrt(r^2+1)
// f = 1 + x^3 * (-6x^2 + 15x - 10), x = 0.2*r  (0 for x >= 1)
__device__ __forceinline__ float chi_ij(float r) {
    const float phi   = __builtin_amdgcn_rsqf(__builtin_fmaf(r, r, 1.0f)); // v_rsq_f32
    const float rcp_r = __builtin_amdgcn_rcpf(r);                          // v_rcp_f32
    const float x  = r * (2.0f / CUTOFF_F);
    const float x3 = x * x * x;
    const float inner = __builtin_fmaf(x, __builtin_fmaf(x, -6.0f, 15.0f), -10.0f);
    float f = __builtin_fmaf(x3, inner, 1.0f);
    f = (x < 1.0f) ? f : 0.0f;
    return __builtin_fmaf(f, phi - rcp_r, rcp_r);  // rcp_r + f*(phi - rcp_r)
}

__device__ __forceinline__ void scatter_add(float* p, float v) {
    (void)__hip_atomic_fetch_add(p, v, __ATOMIC_RELAXED, __HIP_MEMORY_SCOPE_AGENT);
}

// ---------------------------------------------------------------------------
// Zero the output array (harness poisons d_out with 0xAA).
// ---------------------------------------------------------------------------
__global__ void zero_out(float* __restrict__ out, int n) {
    int i = blockIdx.x * blockDim.x + threadIdx.x;
    const int stride = gridDim.x * blockDim.x;
    for (; i < n; i += stride) out[i] = 0.0f;
}

// ---------------------------------------------------------------------------
// Per-molecule charge-neutrality correction (one 128-thread block per mol).
//   corr[m] = (sum(qi[mol m]) - q_ref[m]) / N[m];  qc[a] = qi[a] - corr
// ---------------------------------------------------------------------------
__global__ void correct_q(const float* __restrict__ qi,
                          const float* __restrict__ q_ref,
                          const int*   __restrict__ Ncnt,
                          float*       __restrict__ corr,
                          float*       __restrict__ qc,
                          int apm) {
    __shared__ float red[4];
    const int m    = blockIdx.x;
    const int t    = threadIdx.x;
    const int base = m * apm;

    float v = 0.0f;
    for (int a = t; a < apm; a += blockDim.x) v += qi[base + a];
    #pragma unroll
    for (int off = 16; off > 0; off >>= 1) v += __shfl_down(v, off, 32);  // wave32
    if ((t & 31) == 0) red[t >> 5] = v;
    __syncthreads();
    if (t == 0) {
        const float s = red[0] + red[1] + red[2] + red[3];
        red[0] = (s - q_ref[m]) / (float)Ncnt[m];
    }
    __syncthreads();
    const float c = red[0];
    if (t == 0) corr[m] = c;
    if (qc != nullptr) {
        for (int a = t; a < apm; a += blockDim.x) qc[base + a] = qi[base + a] - c;
    }
}

// ---------------------------------------------------------------------------
// Async-staged edge kernel (main path).
// Tile = 1024 edges per block iteration; wave w owns edges [w*128, w*128+128)
// of the tile. Per tile each wave issues ONE global_load_async_to_lds_b128
// per stream (32 lanes x 16 B = 512 B), double-buffered:
//     issue(t+1) ; s_wait_asynccnt 3 ; consume(t)
// Each wave consumes only its own data -> no cross-wave barriers; ASYNCcnt
// alone orders producer/consumer.
// ---------------------------------------------------------------------------
#define TILE_EDGES   1024
#define WAVE_EDGES   128

__global__ void edge_kernel_async(const float* __restrict__ dist,
                                  const int*   __restrict__ ii,
                                  const int*   __restrict__ jj,
                                  const float* __restrict__ qc,
                                  float*       __restrict__ out,
                                  int n_tiles, int n_main, int n_edges) {
    __shared__ __align__(16) float sm_d[2][8][WAVE_EDGES];
    __shared__ __align__(16) int   sm_i[2][8][WAVE_EDGES];
    __shared__ __align__(16) int   sm_j[2][8][WAVE_EDGES];

    const int w  = threadIdx.x >> 5;   // wave id in block (8 waves)
    const int l  = threadIdx.x & 31;   // lane id (wave32)
    const int l4 = l << 2;             // 4 edges per lane

    // LDS byte offsets for this lane (low 32 bits of generic LDS address
    // carry the LDS offset on CDNA5: aperture maps ADDR[31:0] -> LDS_ADDR).
    const unsigned lds_d0 = (unsigned)(size_t)&sm_d[0][w][l4];
    const unsigned lds_i0 = (unsigned)(size_t)&sm_i[0][w][l4];
    const unsigned lds_j0 = (unsigned)(size_t)&sm_j[0][w][l4];
    const unsigned lds_d1 = (unsigned)(size_t)&sm_d[1][w][l4];
    const unsigned lds_i1 = (unsigned)(size_t)&sm_i[1][w][l4];
    const unsigned lds_j1 = (unsigned)(size_t)&sm_j[1][w][l4];

    const unsigned lane_byte = (unsigned)(w * WAVE_EDGES * 4 + l * 16);

    int tile = blockIdx.x;
    int p = 0;

    if (tile < n_tiles) {
        const unsigned off = (unsigned)tile * (TILE_EDGES * 4) + lane_byte;
        asm volatile("global_load_async_to_lds_b128 %0, %1, %2"
                     :: "v"(lds_d0), "v"(off), "s"(dist) : "memory");
        asm volatile("global_load_async_to_lds_b128 %0, %1, %2"
                     :: "v"(lds_i0), "v"(off), "s"(ii)   : "memory");
        asm volatile("global_load_async_to_lds_b128 %0, %1, %2"
                     :: "v"(lds_j0), "v"(off), "s"(jj)   : "memory");
    }

    for (; tile < n_tiles; tile += gridDim.x) {
        const int  next      = tile + gridDim.x;
        const bool have_next = next < n_tiles;     // uniform across block
        if (have_next) {
            const unsigned off = (unsigned)next * (TILE_EDGES * 4) + lane_byte;
            const unsigned ld = (p == 0) ? lds_d1 : lds_d0;
            const unsigned li = (p == 0) ? lds_i1 : lds_i0;
            const unsigned lj = (p == 0) ? lds_j1 : lds_j0;
            asm volatile("global_load_async_to_lds_b128 %0, %1, %2"
                         :: "v"(ld), "v"(off), "s"(dist) : "memory");
            asm volatile("global_load_async_to_lds_b128 %0, %1, %2"
                         :: "v"(li), "v"(off), "s"(ii)   : "memory");
            asm volatile("global_load_async_to_lds_b128 %0, %1, %2"
                         :: "v"(lj), "v"(off), "s"(jj)   : "memory");
            asm volatile("s_wait_asynccnt 0x3" ::: "memory");  // tile's 3 done
        } else {
            asm volatile("s_wait_asynccnt 0x0" ::: "memory");
        }

        const v4f r = *(const v4f*)&sm_d[p][w][l4];  // ds_load_b128
        const v4i a = *(const v4i*)&sm_i[p][w][l4];
        const v4i b = *(const v4i*)&sm_j[p][w][l4];
        #pragma unroll
        for (int k = 0; k < 4; ++k) {
            const int ia = a[k];
            scatter_add(&out[ia], 0.5f * qc[ia] * qc[b[k]] * chi_ij(r[k]));
        }
        p ^= 1;
    }

    // scalar tail: edges [n_main, n_edges) via direct loads
    const int gstride = gridDim.x * blockDim.x;
    for (int e = n_main + blockIdx.x * blockDim.x + threadIdx.x;
         e < n_edges; e += gstride) {
        const int ia = ii[e];
        scatter_add(&out[ia], 0.5f * qc[ia] * qc[jj[e]] * chi_ij(dist[e]));
    }
}

// ---------------------------------------------------------------------------
// Direct-load edge kernel (fallback: misaligned jj base). b128 NT loads +
// global_prefetch_b8 on the three streams.
// ---------------------------------------------------------------------------
__global__ void edge_kernel_qc(const float* __restrict__ dist,
                               const int*   __restrict__ ii,
                               const int*   __restrict__ jj,
                               const float* __restrict__ qc,
                               float*       __restrict__ out,
                               int n4, int n_edges) {
    const v4f* __restrict__ d4 = (const v4f*)dist;
    const int stride = gridDim.x * blockDim.x;
    const int tid0   = blockIdx.x * blockDim.x + threadIdx.x;

    for (int g = tid0; g < n4; g += stride) {
        __builtin_prefetch((const void*)(d4 + g + stride), 0, 1);
        __builtin_prefetch((const void*)(ii + 4 * (g + stride)), 0, 1);
        __builtin_prefetch((const void*)(jj + 4 * (g + stride)), 0, 1);
        const v4f r = __builtin_nontemporal_load(d4 + g);  // TH=NT streaming
        #pragma unroll
        for (int k = 0; k < 4; ++k) {
            const int ia = __builtin_nontemporal_load(ii + 4 * g + k);
            const int ib = __builtin_nontemporal_load(jj + 4 * g + k);
            scatter_add(&out[ia], 0.5f * qc[ia] * qc[ib] * chi_ij(r[k]));
        }
    }
    for (int e = (n4 << 2) + tid0; e < n_edges; e += stride) {
        const int ia = ii[e];
        scatter_add(&out[ia], 0.5f * qc[ia] * qc[jj[e]] * chi_ij(dist[e]));
    }
}

// Fallback: workspace too small for qc[] -> recompute qi - corr[mol] per edge.
__global__ void edge_kernel_corr(const float* __restrict__ dist,
                                 const int*   __restrict__ ii,
                                 const int*   __restrict__ jj,
                                 const float* __restrict__ qi,
                                 const float* __restrict__ corr,
                                 const int*   __restrict__ mb,
                                 float*       __restrict__ out,
                                 int n_edges) {
    const int stride = gridDim.x * blockDim.x;
    for (int e = blockIdx.x * blockDim.x + threadIdx.x; e < n_edges; e += stride) {
        const int   ia  = ii[e];
        const int   ib  = jj[e];
        const float qia = qi[ia] - corr[mb[ia]];
        const float qib = qi[ib] - corr[mb[ib]];
        scatter_add(&out[ia], 0.5f * qia * qib * chi_ij(dist[e]));
    }
}

extern "C" void kernel_launch(void* const* d_in, const int* in_sizes, int n_in,
                              void* d_out, int out_size, void* d_ws, size_t ws_size,
                              hipStream_t stream) {
    // 0: qi [n_atoms] f32, 1: edge_dist [n_edges] f32,
    // 2: edge_index [2, n_edges] i32, 3: q_ref [n_mol] f32,
    // 4: N [n_mol] i32, 5: atom_mol_batch [n_atoms] i32
    const float* qi    = (const float*)d_in[0];
    const float* dist  = (const float*)d_in[1];
    const int*   eidx  = (const int*)d_in[2];
    const float* q_ref = (const float*)d_in[3];
    const int*   Ncnt  = (const int*)d_in[4];
    const int*   mb    = (const int*)d_in[5];

    const int n_atoms = in_sizes[0];
    const int n_edges = in_sizes[1];
    const int n_mol   = in_sizes[3];
    const int apm     = n_atoms / n_mol;

    float* out  = (float*)d_out;
    float* corr = (float*)d_ws;                       // [n_mol]
    const size_t need_qc = (size_t)n_mol * 4 + (size_t)n_atoms * 4;
    const bool   has_qc  = ws_size >= need_qc;
    float* qc = has_qc ? (corr + n_mol) : nullptr;    // [n_atoms]

    {   // 1) zero output
        int blocks = (n_atoms + 1023) / 1024;
        if (blocks > 2048) blocks = 2048;
        if (blocks < 1) blocks = 1;
        zero_out<<<blocks, 256, 0, stream>>>(out, n_atoms);
    }
    // 2) per-molecule correction
    correct_q<<<n_mol, 128, 0, stream>>>(qi, q_ref, Ncnt, corr, qc, apm);

    // 3) edges
    const int* ii = eidx;
    const int* jj = eidx + n_edges;
    if (has_qc && ((n_edges & 3) == 0)) {
        // async-staged path: needs 16B-aligned jj base (n_edges % 4 == 0)
        const int n_tiles = n_edges / TILE_EDGES;
        const int n_main  = n_tiles * TILE_EDGES;
        int blocks = n_tiles;
        if (blocks > 4096) blocks = 4096;
        if (blocks < 1) blocks = 1;
        edge_kernel_async<<<blocks, 256, 0, stream>>>(dist, ii, jj, qc, out,
                                                      n_tiles, n_main, n_edges);
    } else if (has_qc) {
        const int n4 = n_edges >> 2;
        int blocks = (n4 + 255) / 256;
        if (blocks > 4096) blocks = 4096;
        if (blocks < 1) blocks = 1;
        edge_kernel_qc<<<blocks, 256, 0, stream>>>(dist, ii, jj, qc, out,
                                                   n4, n_edges);
    } else {
        int eb = (n_edges + 255) / 256;
        if (eb > 8192) eb = 8192;
        if (eb < 1) eb = 1;
        edge_kernel_corr<<<eb, 256, 0, stream>>>(dist, ii, jj, qi, corr, mb,
                                                 out, n_edges);
    }
}